// SelectiveQuantizer_5351529251297
// MI455X (gfx1250) — compile-verified
//
#include <hip/hip_runtime.h>
#include <stdint.h>

// SelectiveQuantizer for MI455X (gfx1250): bandwidth-bound fake-quant.
// CDNA5 paths used: global_load_async_to_lds_b128 + s_wait_asynccnt (kernel 1),
// global_prefetch_b8 via __builtin_prefetch, non-temporal b128 stores (kernel 2).

#define IN_F   16384
#define OUT_F  4096
#define NBINS  3

typedef float v4f __attribute__((ext_vector_type(4)));

// ---------------------------------------------------------------------------
// Kernel 1: find sorted_scores[n/3] and sorted_scores[2n/3] via stable ranks.
// Stage all scores (64 KB) into LDS with CDNA5 async-to-LDS loads.
// ---------------------------------------------------------------------------
__global__ __launch_bounds__(256) void sq_thresh_kernel(
    const float* __restrict__ scores, float* __restrict__ thr, int n) {
  __shared__ float s_sc[IN_F];  // 64 KB LDS (WGP has 320 KB)
  const int tid = threadIdx.x;

  // Async-stage scores -> LDS: 16 B per lane per issue, 16 issues covers 64 KB.
  unsigned lds_base = (unsigned)(uintptr_t)(&s_sc[0]);
  #pragma unroll
  for (int it = 0; it < IN_F / 1024; ++it) {
    unsigned off      = (unsigned)(it * 4096 + tid * 16);
    unsigned lds_addr = lds_base + off;
    uint64_t gaddr    = (uint64_t)(uintptr_t)scores + (uint64_t)off;
    asm volatile("global_load_async_to_lds_b128 %0, %1, off"
                 :: "v"(lds_addr), "v"(gaddr) : "memory");
  }
  asm volatile("s_wait_asynccnt 0" ::: "memory");
  __syncthreads();

  const int i = blockIdx.x * blockDim.x + tid;
  if (i >= n) return;
  const float si = s_sc[i];
  int cnt = 0;
  #pragma unroll 8
  for (int j = 0; j < n; ++j) {
    const float sj = s_sc[j];                       // LDS broadcast (uniform j)
    cnt += (sj < si) || (sj == si && j < i);        // stable rank
  }
  const int bin = n / NBINS;
  if (cnt == bin)     thr[0] = si;                  // exactly one writer each
  if (cnt == 2 * bin) thr[1] = si;
}

// ---------------------------------------------------------------------------
// Kernel 2: fused column min/max + params + dequant over a 64-column slab.
// ---------------------------------------------------------------------------
__device__ __forceinline__ float sq_dq(float w, float scale, float zp, int skip) {
  float q = rintf(w / scale) + zp;                  // jnp.round == rne
  q = fminf(fmaxf(q, -128.0f), 127.0f);             // qint8 storage clamp
  const float d = (q - zp) * scale;
  return skip ? w : d;                              // bits==MAX_BITS passthrough
}

__global__ __launch_bounds__(256) void sq_quant_kernel(
    const float* __restrict__ weight, const float* __restrict__ scores,
    const float* __restrict__ thr, float* __restrict__ out) {
  constexpr int COLS = 64;    // columns per block (slab = 1 MB)
  constexpr int RPI  = 16;    // rows in flight per iteration (256 thr / 16 cgs)
  const int tid     = threadIdx.x;
  const int colBase = blockIdx.x * COLS;
  const int cg      = tid & 15;   // col-group: 4 consecutive columns
  const int r0      = tid >> 4;   // row lane 0..15

  __shared__ float aMin[RPI][COLS];
  __shared__ float aMax[RPI][COLS];
  __shared__ float cScale[COLS];
  __shared__ float cZp[COLS];
  __shared__ int   cSkip[COLS];

  // ---- Pass A: stream slab, per-column min/max in registers ----
  v4f mn = {3.402823466e38f, 3.402823466e38f, 3.402823466e38f, 3.402823466e38f};
  v4f mx = {-3.402823466e38f, -3.402823466e38f, -3.402823466e38f, -3.402823466e38f};
  for (int r = r0; r < OUT_F; r += RPI) {
    const size_t base = (size_t)r * IN_F + (size_t)colBase + 4u * (unsigned)cg;
    if (r + 128 < OUT_F)  // global_prefetch_b8, 128 rows ahead
      __builtin_prefetch(weight + base + (size_t)128 * IN_F, 0, 0);
    const v4f v = *reinterpret_cast<const v4f*>(weight + base);
    mn.x = fminf(mn.x, v.x); mn.y = fminf(mn.y, v.y);
    mn.z = fminf(mn.z, v.z); mn.w = fminf(mn.w, v.w);
    mx.x = fmaxf(mx.x, v.x); mx.y = fmaxf(mx.y, v.y);
    mx.z = fmaxf(mx.z, v.z); mx.w = fmaxf(mx.w, v.w);
  }
  aMin[r0][4 * cg + 0] = mn.x; aMin[r0][4 * cg + 1] = mn.y;
  aMin[r0][4 * cg + 2] = mn.z; aMin[r0][4 * cg + 3] = mn.w;
  aMax[r0][4 * cg + 0] = mx.x; aMax[r0][4 * cg + 1] = mx.y;
  aMax[r0][4 * cg + 2] = mx.z; aMax[r0][4 * cg + 3] = mx.w;
  __syncthreads();

  // ---- Per-column parameters (64 lanes) ----
  if (tid < COLS) {
    float cmn = aMin[0][tid], cmx = aMax[0][tid];
    #pragma unroll
    for (int k = 1; k < RPI; ++k) {
      cmn = fminf(cmn, aMin[k][tid]);
      cmx = fmaxf(cmx, aMax[k][tid]);
    }
    const float t0 = thr[0], t1 = thr[1];
    const float s  = scores[colBase + tid];
    // linspace(2,8,4)=[2,4,6,8], bins use indices 0..2 -> bits in {2,4,6}
    const int bits = (s <= t0) ? 2 : ((s <= t1) ? 4 : 6);
    const float qmin = -(float)(1 << (bits - 1));
    const float qmax = (float)((1 << (bits - 1)) - 1);
    float scale = (cmx - cmn) / (qmax - qmin);
    if (fabsf(scale) < 1e-6f) scale = 1e-6f;
    float zp = rintf(qmin - cmn / scale);
    zp = fminf(fmaxf(zp, qmin), qmax);
    cScale[tid] = scale;
    cZp[tid]    = zp;
    cSkip[tid]  = (bits == 8);  // never true here; kept for fidelity
  }
  __syncthreads();

  // ---- Pass B: re-read slab (L2-hot), dequant, non-temporal store ----
  float sc[4], zp4[4];
  int   sk[4];
  #pragma unroll
  for (int k = 0; k < 4; ++k) {
    sc[k]  = cScale[4 * cg + k];
    zp4[k] = cZp[4 * cg + k];
    sk[k]  = cSkip[4 * cg + k];
  }
  for (int r = r0; r < OUT_F; r += RPI) {
    const size_t base = (size_t)r * IN_F + (size_t)colBase + 4u * (unsigned)cg;
    const v4f v = *reinterpret_cast<const v4f*>(weight + base);
    v4f o;
    o.x = sq_dq(v.x, sc[0], zp4[0], sk[0]);
    o.y = sq_dq(v.y, sc[1], zp4[1], sk[1]);
    o.z = sq_dq(v.z, sc[2], zp4[2], sk[2]);
    o.w = sq_dq(v.w, sc[3], zp4[3], sk[3]);
    __builtin_nontemporal_store(o, reinterpret_cast<v4f*>(out + base));
  }
}

// ---------------------------------------------------------------------------
extern "C" void kernel_launch(void* const* d_in, const int* in_sizes, int n_in,
                              void* d_out, int out_size, void* d_ws, size_t ws_size,
                              hipStream_t stream) {
  const float* weight = (const float*)d_in[0];
  const float* scores = (const float*)d_in[1];
  float* out = (float*)d_out;
  float* thr = (float*)d_ws;  // 2 floats of scratch
  const int n = in_sizes[1];  // 16384

  sq_thresh_kernel<<<n / 256, 256, 0, stream>>>(scores, thr, n);
  sq_quant_kernel<<<IN_F / 64, 256, 0, stream>>>(weight, scores, thr, out);
}